// GnnLayer_83562883711167
// MI455X (gfx1250) — compile-verified
//
#include <hip/hip_runtime.h>
#include <hip/hip_bf16.h>

typedef __attribute__((ext_vector_type(2))) float v2f;
typedef __attribute__((ext_vector_type(8))) float v8f;

#define FD 256  // feature dim

// ---------- degree / normalization ----------
__global__ __launch_bounds__(256) void k_init_deg(float* __restrict__ deg, int n) {
    int i = blockIdx.x * 256 + threadIdx.x;
    if (i < n) deg[i] = 1.0f;  // self-loop
}

__global__ __launch_bounds__(256) void k_count_deg(const int* __restrict__ dst,
                                                   float* __restrict__ deg, int e) {
    int i = blockIdx.x * 256 + threadIdx.x;
    if (i < e) atomicAdd(&deg[dst[i]], 1.0f);
}

__global__ __launch_bounds__(256) void k_dinv(float* __restrict__ deg, int n) {
    int i = blockIdx.x * 256 + threadIdx.x;
    if (i < n) deg[i] = rsqrtf(deg[i]);  // deg >= 1 always
}

// ---------- W transpose + K-pair interleave ----------
// W is [out=256, in=256]; B fragment wants lane-local K pairs contiguous:
//   WtP[(k>>1)*512 + n*2 + (k&1)] = W[n][k]
// so a lane loads its (K, K+1) pair for column n as a single 8-byte load.
__global__ __launch_bounds__(256) void k_transpose(const float* __restrict__ W,
                                                   float* __restrict__ WtP) {
    int i = blockIdx.x * 256 + threadIdx.x;  // i enumerates (k, n)
    int k = i >> 8, n = i & 255;
    WtP[(k >> 1) * 512 + n * 2 + (k & 1)] = W[n * FD + k];
}

// ---------- agg init with self-loop term: agg = x[n] * dinv[n]^2 ----------
__global__ __launch_bounds__(256) void k_init_agg(const float* __restrict__ nodes,
                                                  const float* __restrict__ dinv,
                                                  float* __restrict__ agg, int n) {
    long long i = (long long)blockIdx.x * 256 + threadIdx.x;
    if (i < (long long)n * FD) {
        int row = (int)(i >> 8);
        float di = dinv[row];
        agg[i] = nodes[i] * di * di;
    }
}

// ---------- edge scatter: one wave per edge, coalesced atomics ----------
__global__ __launch_bounds__(256) void k_scatter(const float* __restrict__ nodes,
                                                 const int* __restrict__ src,
                                                 const int* __restrict__ dst,
                                                 const float* __restrict__ dinv,
                                                 float* __restrict__ agg, int e) {
    int wv = (blockIdx.x * 256 + threadIdx.x) >> 5;
    int lane = threadIdx.x & 31;
    if (wv >= e) return;
    int s = src[wv], d = dst[wv];
    float coeff = dinv[s] * dinv[d];
    const float* xs = nodes + (size_t)s * FD;
    float* od = agg + (size_t)d * FD;
#pragma unroll
    for (int j = 0; j < 8; ++j) {
        int c = lane + j * 32;
        atomicAdd(od + c, xs[c] * coeff);
    }
}

// ---------- fused GEMM (agg @ W^T) + bias + LayerNorm + ReLU ----------
// block = 256 threads (8 waves), computes 16 output rows x 256 cols.
// agg may alias out: each block reads only its own 16 rows (staged to LDS first)
// and writes only those rows afterwards.
__global__ __launch_bounds__(256) void k_gemm_ln(const float* __restrict__ agg,
                                                 const float* __restrict__ WtP,
                                                 const float* __restrict__ bias,
                                                 const float* __restrict__ gamma,
                                                 const float* __restrict__ beta,
                                                 float* __restrict__ out, int n) {
    __shared__ float S[16 * 260];  // padded stride 260 (8B-aligned rows, spread banks)
    const int tid = threadIdx.x;
    const int wave = tid >> 5;
    const int lane = tid & 31;
    const int rowBase = blockIdx.x * 16;

    // stage A tile (16 x 256 f32) into LDS, fully coalesced float4 loads
#pragma unroll
    for (int rep = 0; rep < 4; ++rep) {
        int flat = rep * 1024 + tid * 4;
        int r = flat >> 8;
        int c = flat & 255;
        float4 v = make_float4(0.f, 0.f, 0.f, 0.f);
        if (rowBase + r < n)
            v = *reinterpret_cast<const float4*>(agg + (size_t)(rowBase + r) * FD + c);
        float* p = &S[r * 260 + c];
        p[0] = v.x; p[1] = v.y; p[2] = v.z; p[3] = v.w;
    }
    __syncthreads();

    const int m = lane & 15;     // A row / B col within tile
    const int half = lane >> 4;  // lane group selects K pair
    const int col0 = wave * 32;  // this wave owns cols [col0, col0+32)

    v8f acc0 = {};
    v8f acc1 = {};
    for (int k0 = 0; k0 < FD; k0 += 4) {
        const int ka = k0 + half * 2;  // even: (ka, ka+1) is one K pair
        v2f a;
        a[0] = S[m * 260 + ka];
        a[1] = S[m * 260 + ka + 1];
        // packed B: one 8-byte load per fragment, coalesced across lanes
        const float2* wp = reinterpret_cast<const float2*>(WtP + (size_t)(ka >> 1) * 512);
        float2 f0 = wp[col0 + m];
        float2 f1 = wp[col0 + 16 + m];
        v2f b0; b0[0] = f0.x; b0[1] = f0.y;
        v2f b1; b1[0] = f1.x; b1[1] = f1.y;
        acc0 = __builtin_amdgcn_wmma_f32_16x16x4_f32(false, a, false, b0, (short)0,
                                                     acc0, false, false);
        acc1 = __builtin_amdgcn_wmma_f32_16x16x4_f32(false, a, false, b1, (short)0,
                                                     acc1, false, false);
    }
    __syncthreads();  // all waves done reading A tile; reuse LDS for output tile

    // C layout: VGPR r holds row (r + 8*half), col = col0 + (lane&15)
#pragma unroll
    for (int r = 0; r < 8; ++r) {
        int row = r + half * 8;
        int c = col0 + m;
        S[row * 260 + c] = acc0[r] + bias[c];
        S[row * 260 + c + 16] = acc1[r] + bias[c + 16];
    }
    __syncthreads();

    // LayerNorm + ReLU: wave w handles rows {2w, 2w+1}
#pragma unroll
    for (int rr = 0; rr < 2; ++rr) {
        int row = wave * 2 + rr;
        float s1 = 0.f, s2 = 0.f;
        float vals[8];
#pragma unroll
        for (int j = 0; j < 8; ++j) {
            float v = S[row * 260 + lane + j * 32];
            vals[j] = v;
            s1 += v;
            s2 += v * v;
        }
        for (int off = 16; off > 0; off >>= 1) {
            s1 += __shfl_xor(s1, off, 32);
            s2 += __shfl_xor(s2, off, 32);
        }
        float mu = s1 * (1.0f / FD);
        float var = s2 * (1.0f / FD) - mu * mu;
        float inv = rsqrtf(var + 1e-5f);
        if (rowBase + row < n) {
            float* orow = out + (size_t)(rowBase + row) * FD;
#pragma unroll
            for (int j = 0; j < 8; ++j) {
                int c = lane + j * 32;
                float v = (vals[j] - mu) * inv * gamma[c] + beta[c];
                orow[c] = fmaxf(v, 0.0f);
            }
        }
    }
}

extern "C" void kernel_launch(void* const* d_in, const int* in_sizes, int n_in,
                              void* d_out, int out_size, void* d_ws, size_t ws_size,
                              hipStream_t stream) {
    const float* nodes = (const float*)d_in[0];
    const int* ei      = (const int*)d_in[1];   // [2, E] flat: src row then dst row
    const float* W     = (const float*)d_in[2];
    const float* bias  = (const float*)d_in[3];
    const float* gamma = (const float*)d_in[4];
    const float* beta  = (const float*)d_in[5];
    float* out = (float*)d_out;

    const int N = in_sizes[0] / FD;
    const int E = in_sizes[1] / 2;
    const int* src = ei;
    const int* dst = ei + E;

    // workspace: dinv[N] then packed-transposed W (< 1 MB total)
    char* ws = (char*)d_ws;
    float* dinv = (float*)ws;
    size_t offWt = ((size_t)N * sizeof(float) + 255) & ~(size_t)255;
    float* WtP = (float*)(ws + offWt);

    // agg accumulates directly in d_out (each GEMM block reads its own rows
    // into LDS before overwriting them, so aliasing is safe).
    float* agg = out;

    // 1) W transpose + K-pair pack (independent)
    k_transpose<<<(FD * FD) / 256, 256, 0, stream>>>(W, WtP);
    // 2) degree with self-loop, then rsqrt
    k_init_deg<<<(N + 255) / 256, 256, 0, stream>>>(dinv, N);
    k_count_deg<<<(E + 255) / 256, 256, 0, stream>>>(dst, dinv, E);
    k_dinv<<<(N + 255) / 256, 256, 0, stream>>>(dinv, N);
    // 3) agg = self-loop term
    {
        long long total = (long long)N * FD;
        int blocks = (int)((total + 255) / 256);
        k_init_agg<<<blocks, 256, 0, stream>>>(nodes, dinv, agg, N);
    }
    // 4) edge scatter: one wave per edge
    {
        long long threads = (long long)E * 32;
        int blocks = (int)((threads + 255) / 256);
        k_scatter<<<blocks, 256, 0, stream>>>(nodes, src, dst, dinv, agg, E);
    }
    // 5) fused WMMA GEMM + bias + LayerNorm + ReLU
    k_gemm_ln<<<(N + 15) / 16, 256, 0, stream>>>(agg, WtP, bias, gamma, beta, out, N);
}